// CapsuleLayer_61297773248626
// MI455X (gfx1250) — compile-verified
//
#include <hip/hip_runtime.h>

typedef __attribute__((ext_vector_type(2))) float v2f;
typedef __attribute__((ext_vector_type(4))) float v4f;
typedef __attribute__((ext_vector_type(8))) float v8f;

#define J_CAPS   10
#define D_CAP    16
#define CHANNELS 32
#define IN_NUM   6400
#define IN_DIM   8
#define BATCH    128
#define REP      (IN_NUM / CHANNELS)   /* 200 i-columns per channel */
#define TILES_PER_CH 13                /* ceil(200/16); last tile has 8 valid cols */
#define WAVES    8
#define BLOCK    (WAVES * 32)

__global__ __launch_bounds__(BLOCK)
void capsule_routing_kernel(const float* __restrict__ X,    // [B, I, K]
                            const float* __restrict__ W,    // [J, C, D, K]
                            const float* __restrict__ Bias, // [J, D]
                            float* __restrict__ out)        // [B, J, D]
{
    extern __shared__ float lds_x[];                           // IN_NUM*IN_DIM floats (204.8 KB)
    __shared__ __align__(16) float s_slots[WAVES][J_CAPS][D_CAP];
    __shared__ __align__(16) float s_accumO[J_CAPS][D_CAP];
    __shared__ __align__(16) float s_v[J_CAPS][D_CAP];
    __shared__ __align__(16) float g_all[CHANNELS][J_CAPS][IN_DIM];  // G per channel (10 KB)

    const int b    = blockIdx.x;
    const int tid  = threadIdx.x;
    const int lane = tid & 31;
    const int wave = tid >> 5;
    const int half = lane >> 4;       // 0: lanes 0-15, 1: lanes 16-31
    const int l16  = lane & 15;
    const int k0   = half * 2;        // 32-bit frag: VGPR0 holds K=k0, VGPR1 holds K=k0+1

    // ---- stage inputs[b] into LDS (vectorized) ----
    {
        const v4f* gx  = (const v4f*)(X + (size_t)b * IN_NUM * IN_DIM);
        v4f*       lx4 = (v4f*)lds_x;
        const int n4 = IN_NUM * IN_DIM / 4;          // 12800
        for (int idx = tid; idx < n4; idx += BLOCK) lx4[idx] = gx[idx];
    }
    if (tid < J_CAPS * D_CAP) ((float*)s_accumO)[tid] = 0.0f;
    __syncthreads();

    for (int pass = 0; pass < 3; ++pass) {
        // ---- build G[ch][j][k] = sum_d accumO[j][d] * W[j,ch,d,k] for ALL channels ----
#pragma unroll 1
        for (int task = tid; task < J_CAPS * CHANNELS; task += BLOCK) {
            const int j  = task / CHANNELS;
            const int ch = task % CHANNELS;
            const v4f* wp = (const v4f*)(W + ((size_t)j * CHANNELS + ch) * D_CAP * IN_DIM);
            v4f acc0 = {0.0f, 0.0f, 0.0f, 0.0f};
            v4f acc1 = {0.0f, 0.0f, 0.0f, 0.0f};
#pragma unroll
            for (int d = 0; d < D_CAP; ++d) {
                const float a = s_accumO[j][d];
                acc0 += wp[2 * d]     * a;
                acc1 += wp[2 * d + 1] * a;
            }
            *(v4f*)&g_all[ch][j][0] = acc0;
            *(v4f*)&g_all[ch][j][4] = acc1;
        }
        __syncthreads();

        // partS8[j]: WMMA C/D accumulator, S[j][d = v + 8*half] summed over i-columns
        v8f partS8[J_CAPS];
#pragma unroll
        for (int j = 0; j < J_CAPS; ++j) partS8[j] = (v8f){};

#pragma unroll 1
        for (int cq = 0; cq < CHANNELS / WAVES; ++cq) {
            const int ch = wave + cq * WAVES;

            // A fragments (weights): A[j] = W[j,ch] as [16 d x 8 k]
            v2f Af[J_CAPS][2];
#pragma unroll
            for (int j = 0; j < J_CAPS; ++j) {
                const float* wp = W + (((size_t)j * CHANNELS + ch) * D_CAP + l16) * IN_DIM;
                Af[j][0] = *(const v2f*)(wp + k0);        // K = k0, k0+1
                Af[j][1] = *(const v2f*)(wp + k0 + 4);    // K = k0+4, k0+5
            }
            // G fragment: [16 j x 8 k], rows 10..15 zero
            const int   jr  = (l16 < J_CAPS) ? l16 : 0;
            const float msk = (l16 < J_CAPS) ? 1.0f : 0.0f;
            const v2f Gf0 = *(const v2f*)&g_all[ch][jr][k0]     * msk;
            const v2f Gf1 = *(const v2f*)&g_all[ch][jr][k0 + 4] * msk;

            const int rowbase = ch * REP;
#pragma unroll 1
            for (int t = 0; t < TILES_PER_CH; ++t) {
                // B fragments (inputs^T): col N = i = rowbase + t*16 + l16
                int irow = rowbase + t * 16 + l16;
                irow = irow < (IN_NUM - 1) ? irow : (IN_NUM - 1); // clamp tail (masked below)
                const float* xp = lds_x + irow * IN_DIM;
                v2f X0 = *(const v2f*)(xp + k0);
                v2f X1 = *(const v2f*)(xp + k0 + 4);

                // ---- phase A: logit tile = G x x^T : [16 j x 16 i], lane=i, rows j ----
                v8f z  = (v8f){};
                v8f l0 = __builtin_amdgcn_wmma_f32_16x16x4_f32(
                    false, Gf0, false, X0, (short)0, z, false, false);
                v8f lt = __builtin_amdgcn_wmma_f32_16x16x4_f32(
                    false, Gf1, false, X1, (short)0, l0, false, false);

                // softmax over j: half0 rows j=0..7, half1 rows j=8..9 (rest excluded)
                const float m_all = fmaxf(fmaxf(fmaxf(lt[0], lt[1]), fmaxf(lt[2], lt[3])),
                                          fmaxf(fmaxf(lt[4], lt[5]), fmaxf(lt[6], lt[7])));
                const float m_2   = fmaxf(lt[0], lt[1]);
                float mm = half ? m_2 : m_all;
                mm = fmaxf(mm, __shfl_xor(mm, 16, 32));

                float e[8];
#pragma unroll
                for (int v = 0; v < 8; ++v) e[v] = __expf(lt[v] - mm);
                const float s_all = ((e[0] + e[1]) + (e[2] + e[3])) +
                                    ((e[4] + e[5]) + (e[6] + e[7]));
                const float s_2 = e[0] + e[1];
                float ss = half ? s_2 : s_all;
                ss += __shfl_xor(ss, 16, 32);
                // tail tile: cols l16 >= 8 belong to the next channel -> zero their c
                const bool valid = (t != TILES_PER_CH - 1) || (l16 < (REP - (TILES_PER_CH - 1) * 16));
                const float inv = valid ? (1.0f / ss) : 0.0f;

                // partner exchange: assemble c_j for all 10 j in every lane
                float ex[8];
#pragma unroll
                for (int v = 0; v < 8; ++v) ex[v] = __shfl_xor(e[v], 16, 32);
                float cj[J_CAPS];
#pragma unroll
                for (int j = 0; j < 8; ++j) cj[j] = half ? ex[j] : e[j];
                cj[8] = half ? e[0] : ex[0];
                cj[9] = half ? e[1] : ex[1];

                // ---- phase C: S accumulation on the matrix pipe ----
#pragma unroll
                for (int j = 0; j < J_CAPS; ++j) {
                    const float c = cj[j] * inv;
                    const v2f XS0 = X0 * c;
                    const v2f XS1 = X1 * c;
                    v8f t0 = __builtin_amdgcn_wmma_f32_16x16x4_f32(
                        false, Af[j][0], false, XS0, (short)0, partS8[j], false, false);
                    partS8[j] = __builtin_amdgcn_wmma_f32_16x16x4_f32(
                        false, Af[j][1], false, XS1, (short)0, t0, false, false);
                }
            }
        }

        // reduce partS over the 16 i-lanes of each half (butterfly within 16-group)
#pragma unroll
        for (int j = 0; j < J_CAPS; ++j)
#pragma unroll
            for (int v = 0; v < 8; ++v) {
                float p = partS8[j][v];
                p += __shfl_xor(p, 1, 32);
                p += __shfl_xor(p, 2, 32);
                p += __shfl_xor(p, 4, 32);
                p += __shfl_xor(p, 8, 32);
                partS8[j][v] = p;
            }
        // lane 0 of each half publishes d=0..7 / d=8..15 of this wave's partial
        if (l16 == 0) {
#pragma unroll
            for (int j = 0; j < J_CAPS; ++j)
#pragma unroll
                for (int v = 0; v < 8; ++v) s_slots[wave][j][v + 8 * half] = partS8[j][v];
        }
        __syncthreads();

        // wave 0: reduce slots, add bias, squash, update accumO
        if (wave == 0) {
#pragma unroll
            for (int j = 0; j < J_CAPS; ++j) {
                float s = 0.0f;
#pragma unroll
                for (int w = 0; w < WAVES; ++w) s += s_slots[w][j][l16];
                s += Bias[j * D_CAP + l16];
                float sq = s * s;
                sq += __shfl_xor(sq, 1, 32);
                sq += __shfl_xor(sq, 2, 32);
                sq += __shfl_xor(sq, 4, 32);
                sq += __shfl_xor(sq, 8, 32);
                const float scale = sq / ((1.0f + sq) * __fsqrt_rn(sq));
                const float vv = scale * s;
                if (half == 0) {
                    s_v[j][l16] = vv;
                    s_accumO[j][l16] += vv;   // b_{t+1} logits = accumO . u_hat
                }
            }
        }
        __syncthreads();
    }

    // final outputs: [b, j, d]
    if (tid < J_CAPS * D_CAP) {
        const int j = tid / D_CAP, d = tid % D_CAP;
        out[((size_t)b * J_CAPS + j) * D_CAP + d] = s_v[j][d];
    }
}

extern "C" void kernel_launch(void* const* d_in, const int* in_sizes, int n_in,
                              void* d_out, int out_size, void* d_ws, size_t ws_size,
                              hipStream_t stream) {
    (void)in_sizes; (void)n_in; (void)out_size; (void)d_ws; (void)ws_size;
    const float* X    = (const float*)d_in[0];  // [128, 6400, 8]
    const float* W    = (const float*)d_in[1];  // [10, 32, 16, 8]
    const float* Bias = (const float*)d_in[2];  // [10, 16]
    float* out        = (float*)d_out;          // [128, 10, 16]

    const size_t smem = (size_t)IN_NUM * IN_DIM * sizeof(float); // 204.8 KB dynamic LDS
    (void)hipFuncSetAttribute((const void*)capsule_routing_kernel,
                              hipFuncAttributeMaxDynamicSharedMemorySize, (int)smem);
    capsule_routing_kernel<<<BATCH, BLOCK, smem, stream>>>(X, W, Bias, out);
}